// GCNConvEmbedding_54932631715890
// MI455X (gfx1250) — compile-verified
//
#include <hip/hip_runtime.h>
#include <hip/hip_bf16.h>

typedef __attribute__((ext_vector_type(2))) float v2f;
typedef __attribute__((ext_vector_type(8))) float v8f;

#define D 64  // feature width for all layers

// ---------------- degree / normalization ----------------

__global__ void deg_init_kernel(float* __restrict__ deg, int n) {
    int i = blockIdx.x * blockDim.x + threadIdx.x;
    if (i < n) deg[i] = 1.0f;  // self-loop weight
}

__global__ void deg_accum_kernel(const int* __restrict__ dst,
                                 const float* __restrict__ w,
                                 float* __restrict__ deg, int E) {
    int e = blockIdx.x * blockDim.x + threadIdx.x;
    if (e < E) atomicAdd(&deg[dst[e]], w[e]);
}

__global__ void dinv_kernel(float* __restrict__ deg, int n) {
    int i = blockIdx.x * blockDim.x + threadIdx.x;
    if (i < n) {
        float d = deg[i];
        deg[i] = (d > 0.0f) ? rsqrtf(fmaxf(d, 1e-30f)) : 0.0f;
    }
}

// ---------------- dense GEMM via V_WMMA_F32_16X16X4_F32 ----------------
// H[n x 64] = X[n x 64] @ W[64 x 64]
// One wave computes one 16x16 output tile; 8 waves/block -> 32 rows x 64 cols.
__global__ void gemm_wmma_kernel(const float* __restrict__ X,
                                 const float* __restrict__ W,
                                 float* __restrict__ H, int n) {
    const int wave = threadIdx.x >> 5;          // 0..7
    const int lane = threadIdx.x & 31;
    const int rowTile = blockIdx.x * 2 + (wave >> 2);
    const int colTile = wave & 3;               // 4 col tiles of 16 -> 64 cols
    const int r0 = rowTile * 16;
    const int c0 = colTile * 16;
    const int half = lane >> 4;                 // 0: K={0,1}, 1: K={2,3}
    const int l = lane & 15;

    int rowA = r0 + l;
    if (rowA > n - 1) rowA = n - 1;             // clamp loads; store is guarded
    const float* __restrict__ xrow = X + (size_t)rowA * D + half * 2;
    const float* __restrict__ wcol = W + (c0 + l) + half * 2 * D;

    v8f acc = {};
#pragma unroll
    for (int k = 0; k < D; k += 4) {
        v2f a, b;
        a.x = xrow[k];
        a.y = xrow[k + 1];
        b.x = wcol[k * D];
        b.y = wcol[k * D + D];
        acc = __builtin_amdgcn_wmma_f32_16x16x4_f32(
            /*neg_a=*/false, a, /*neg_b=*/false, b,
            /*c_mod=*/(short)0, acc, /*reuse_a=*/false, /*reuse_b=*/false);
    }

    // C/D layout: VGPR v, lanes 0-15 -> row r0+v; lanes 16-31 -> row r0+v+8
    float* __restrict__ hp = H + (size_t)(r0 + half * 8) * D + c0 + l;
#pragma unroll
    for (int v = 0; v < 8; ++v) {
        int row = r0 + half * 8 + v;
        if (row < n) hp[(size_t)v * D] = acc[v];
    }
}

// ---------------- aggregation ----------------
// agg[i][f] = bias[f] + h[i][f] * dinv[i]^2   (self-loop message, norm = dinv*1*dinv)
__global__ void init_agg_kernel(const float* __restrict__ h,
                                const float* __restrict__ dinv,
                                const float* __restrict__ bias,
                                float* __restrict__ agg, int n) {
    int idx = blockIdx.x * blockDim.x + threadIdx.x;
    int total = n * D;
    if (idx < total) {
        int i = idx >> 6;
        int f = idx & (D - 1);
        float di = dinv[i];
        agg[idx] = bias[f] + h[idx] * di * di;
    }
}

// One wave per edge; lane handles features 2*lane, 2*lane+1.
__global__ void edge_scatter_kernel(const float* __restrict__ H,
                                    const int* __restrict__ src,
                                    const int* __restrict__ dst,
                                    const float* __restrict__ w,
                                    const float* __restrict__ dinv,
                                    float* __restrict__ agg, int E) {
    int gid = blockIdx.x * blockDim.x + threadIdx.x;
    int e = gid >> 5;
    int lane = gid & 31;
    if (e >= E) return;
    int s = src[e];
    int d = dst[e];
    float norm = dinv[s] * w[e] * dinv[d];
    const float2* __restrict__ hp =
        reinterpret_cast<const float2*>(H + (size_t)s * D);
    float2 hv = hp[lane];
    float* __restrict__ ap = agg + (size_t)d * D + lane * 2;
    atomicAdd(ap, hv.x * norm);
    atomicAdd(ap + 1, hv.y * norm);
}

__global__ void relu_kernel(float* __restrict__ p, int total) {
    int idx = blockIdx.x * blockDim.x + threadIdx.x;
    if (idx < total) p[idx] = fmaxf(p[idx], 0.0f);
}

// ---------------- launch ----------------

extern "C" void kernel_launch(void* const* d_in, const int* in_sizes, int n_in,
                              void* d_out, int out_size, void* d_ws, size_t ws_size,
                              hipStream_t stream) {
    (void)n_in; (void)out_size; (void)ws_size;

    const float* x  = (const float*)d_in[0];
    const int*   ei = (const int*)  d_in[1];
    const float* ea = (const float*)d_in[2];
    const float* W1 = (const float*)d_in[3];
    const float* b1 = (const float*)d_in[4];
    const float* W2 = (const float*)d_in[5];
    const float* b2 = (const float*)d_in[6];
    const float* W3 = (const float*)d_in[7];
    const float* b3 = (const float*)d_in[8];

    const int N = in_sizes[0] / D;
    const int E = in_sizes[2];
    const int* src = ei;
    const int* dst = ei + E;

    float* out  = (float*)d_out;
    float* dinv = (float*)d_ws;                  // N floats (deg -> dinv in place)
    float* bufA = dinv + N;                      // N*D floats (GEMM output h)
    float* bufB = bufA + (size_t)N * D;          // N*D floats (agg / next input)

    const int total = N * D;
    const dim3 blk(256);
    const dim3 gN((N + 255) / 256);
    const dim3 gE((E + 255) / 256);
    const dim3 gT((total + 255) / 256);
    const dim3 gGemm((N + 31) / 32);
    const dim3 gEdge((E + 7) / 8);               // 8 waves (edges) per 256-thr block

    // normalization: deg = 1 + segment_sum(w, dst); dinv = rsqrt(deg)
    deg_init_kernel<<<gN, blk, 0, stream>>>(dinv, N);
    deg_accum_kernel<<<gE, blk, 0, stream>>>(dst, ea, dinv, E);
    dinv_kernel<<<gN, blk, 0, stream>>>(dinv, N);

    // layer 1: x -> bufA (gemm) -> bufB (agg) -> relu
    gemm_wmma_kernel<<<gGemm, blk, 0, stream>>>(x, W1, bufA, N);
    init_agg_kernel<<<gT, blk, 0, stream>>>(bufA, dinv, b1, bufB, N);
    edge_scatter_kernel<<<gEdge, blk, 0, stream>>>(bufA, src, dst, ea, dinv, bufB, E);
    relu_kernel<<<gT, blk, 0, stream>>>(bufB, total);

    // layer 2: bufB -> bufA (gemm) -> bufB (agg, overwrites old h) -> relu
    gemm_wmma_kernel<<<gGemm, blk, 0, stream>>>(bufB, W2, bufA, N);
    init_agg_kernel<<<gT, blk, 0, stream>>>(bufA, dinv, b2, bufB, N);
    edge_scatter_kernel<<<gEdge, blk, 0, stream>>>(bufA, src, dst, ea, dinv, bufB, E);
    relu_kernel<<<gT, blk, 0, stream>>>(bufB, total);

    // layer 3: bufB -> bufA (gemm) -> d_out (agg) -> relu
    gemm_wmma_kernel<<<gGemm, blk, 0, stream>>>(bufB, W3, bufA, N);
    init_agg_kernel<<<gT, blk, 0, stream>>>(bufA, dinv, b3, out, N);
    edge_scatter_kernel<<<gEdge, blk, 0, stream>>>(bufA, src, dst, ea, dinv, out, E);
    relu_kernel<<<gT, blk, 0, stream>>>(out, total);
}